// InputLayer_36833639530665
// MI455X (gfx1250) — compile-verified
//
#include <hip/hip_runtime.h>
#include <stdint.h>

// ---------------------------------------------------------------------------
// Dense-voxel-grid pooling for 128^3 grid (21-bit compact keys).
// Key order (x<<14)|(y<<7)|z == reference order (x*4096+y)*4096+z.
// ---------------------------------------------------------------------------

#define KEYS_BITS 21
#define NKEYS (1u << KEYS_BITS)          // 2,097,152 voxel slots = 128^3
#define TPB 256
#define SCAN_ELEMS 8
#define SCAN_TILE (TPB * SCAN_ELEMS)     // 2048 keys per scan block
#define SCAN_NBLK (NKEYS / SCAN_TILE)    // 1024 scan blocks

#define ACC_TPB 256
#define TILE_DW (3 * ACC_TPB)            // 768 dwords (3 KB) per tile per array
#define ACC_GRID 4096                    // persistent blocks for k_accum

__device__ __forceinline__ uint32_t voxel_key(float x, float y, float z) {
    int cx = (int)floorf(x);
    int cy = (int)floorf(y);
    int cz = (int)floorf(z);
    cx = cx < 0 ? 0 : (cx > 127 ? 127 : cx);
    cy = cy < 0 ? 0 : (cy > 127 ? 127 : cy);
    cz = cz < 0 ? 0 : (cz > 127 ? 127 : cz);
    return ((uint32_t)cx << 14) | ((uint32_t)cy << 7) | (uint32_t)cz;
}

__device__ __forceinline__ unsigned lds_off(const void* p) {
    // generic LDS address: low 32 bits are the workgroup-relative LDS offset
    return (unsigned)(unsigned long long)p;
}

// --- 1. clear the accumulators (must run every call) -----------------------
__global__ __launch_bounds__(TPB) void k_init(uint32_t* __restrict__ count,
                                              uint32_t* __restrict__ firstIdx,
                                              float* __restrict__ sums) {
    uint32_t k = blockIdx.x * TPB + threadIdx.x;
    if (k < NKEYS) {
        count[k]    = 0u;
        firstIdx[k] = 0xFFFFFFFFu;
        sums[3u * k + 0u] = 0.0f;
        sums[3u * k + 1u] = 0.0f;
        sums[3u * k + 2u] = 0.0f;
    }
}

// --- 2. scatter-accumulate: persistent blocks, double-buffered async->LDS --
__global__ __launch_bounds__(ACC_TPB) void k_accum(const float* __restrict__ pos,
                                                   const float* __restrict__ feat,
                                                   int n, int ntiles,
                                                   uint32_t* __restrict__ count,
                                                   uint32_t* __restrict__ firstIdx,
                                                   float* __restrict__ sums,
                                                   uint32_t* __restrict__ keys) {
    __shared__ float sp[2][TILE_DW];   // staged positions
    __shared__ float sf[2][TILE_DW];   // staged features
    const int t = threadIdx.x;
    const long total = (long)n * 3;

    // Stage one tile into LDS buffer `buf`: exactly 6 async instructions per
    // wave (uniform; tail addresses are clamped in-bounds so EXEC never
    // partially masks an issue and the per-wave ASYNCcnt bookkeeping is exact).
    auto stage = [&](int tile, int buf) {
        const long gbase = (long)tile * TILE_DW;
#pragma unroll
        for (int j = 0; j < 3; ++j) {
            long gi = gbase + t + j * ACC_TPB;
            if (gi >= total) gi = total - 1;           // clamp, value unused
            unsigned lp = lds_off(&sp[buf][t + j * ACC_TPB]);
            unsigned lf = lds_off(&sf[buf][t + j * ACC_TPB]);
            const float* gp = pos + gi;
            const float* gf = feat + gi;
            asm volatile("global_load_async_to_lds_b32 %0, %1, off"
                         :: "v"(lp), "v"(gp) : "memory");
            asm volatile("global_load_async_to_lds_b32 %0, %1, off"
                         :: "v"(lf), "v"(gf) : "memory");
        }
    };

    int tile = blockIdx.x;
    if (tile >= ntiles) return;                        // uniform per block
    int buf = 0;
    stage(tile, 0);

    for (; tile < ntiles; tile += gridDim.x) {
        const int next = tile + gridDim.x;
        const bool have_next = next < ntiles;          // uniform per block
        if (have_next) {
            stage(next, buf ^ 1);
            // far-ahead L2 prefetch of the tile after next (speculative)
            __builtin_prefetch(pos  + (long)(tile + 2 * gridDim.x) * TILE_DW + t, 0, 2);
            __builtin_prefetch(feat + (long)(tile + 2 * gridDim.x) * TILE_DW + t, 0, 2);
            // in-order completion: count<=6 => current tile's 6 ops are done
            asm volatile("s_wait_asynccnt 6" ::: "memory");
        } else {
            asm volatile("s_wait_asynccnt 0" ::: "memory");
        }
        __syncthreads();                               // cross-wave visibility

        const int i = tile * ACC_TPB + t;
        if (i < n) {
            float x = sp[buf][3 * t + 0];
            float y = sp[buf][3 * t + 1];
            float z = sp[buf][3 * t + 2];
            uint32_t key = voxel_key(x, y, z);
            atomicAdd(&count[key], 1u);
            atomicMin(&firstIdx[key], (uint32_t)i);
            atomicAdd(&sums[3u * key + 0u], sf[buf][3 * t + 0]);
            atomicAdd(&sums[3u * key + 1u], sf[buf][3 * t + 1]);
            atomicAdd(&sums[3u * key + 2u], sf[buf][3 * t + 2]);
            if (keys) keys[i] = key;
        }
        __syncthreads();                               // buf reusable next iter
        buf ^= 1;
    }
}

// --- 3a. per-block occupancy totals ----------------------------------------
__global__ __launch_bounds__(TPB) void k_scan1(const uint32_t* __restrict__ count,
                                               uint32_t* __restrict__ blockSums) {
    __shared__ uint32_t sm[TPB];
    uint32_t base = blockIdx.x * SCAN_TILE + threadIdx.x * SCAN_ELEMS;
    uint32_t s = 0;
#pragma unroll
    for (int e = 0; e < SCAN_ELEMS; ++e) s += (count[base + e] != 0u) ? 1u : 0u;
    sm[threadIdx.x] = s;
    __syncthreads();
    for (int off = TPB / 2; off > 0; off >>= 1) {
        if ((int)threadIdx.x < off) sm[threadIdx.x] += sm[threadIdx.x + off];
        __syncthreads();
    }
    if (threadIdx.x == 0) blockSums[blockIdx.x] = sm[0];
}

// --- 3b. exclusive scan of the 1024 block totals (single workgroup) --------
__global__ __launch_bounds__(SCAN_NBLK) void k_scan2(uint32_t* __restrict__ blockSums) {
    __shared__ uint32_t sa[SCAN_NBLK];
    __shared__ uint32_t sb[SCAN_NBLK];
    int t = threadIdx.x;
    sa[t] = blockSums[t];
    __syncthreads();
    uint32_t* src = sa;
    uint32_t* dst = sb;
    for (int off = 1; off < SCAN_NBLK; off <<= 1) {
        uint32_t v = src[t];
        if (t >= off) v += src[t - off];
        dst[t] = v;
        __syncthreads();
        uint32_t* tmp = src; src = dst; dst = tmp;
    }
    blockSums[t] = (t == 0) ? 0u : src[t - 1];   // exclusive
}

// --- 3c. intra-block exclusive scan -> global rank per voxel slot ----------
__global__ __launch_bounds__(TPB) void k_scan3(const uint32_t* __restrict__ count,
                                               const uint32_t* __restrict__ blockSums,
                                               uint32_t* __restrict__ rank) {
    __shared__ uint32_t sa[TPB];
    __shared__ uint32_t sb[TPB];
    int t = threadIdx.x;
    uint32_t base = blockIdx.x * SCAN_TILE + t * SCAN_ELEMS;
    uint32_t occ[SCAN_ELEMS];
    uint32_t s = 0;
#pragma unroll
    for (int e = 0; e < SCAN_ELEMS; ++e) {
        occ[e] = (count[base + e] != 0u) ? 1u : 0u;
        s += occ[e];
    }
    sa[t] = s;
    __syncthreads();
    uint32_t* src = sa;
    uint32_t* dst = sb;
    for (int off = 1; off < TPB; off <<= 1) {
        uint32_t v = src[t];
        if (t >= off) v += src[t - off];
        dst[t] = v;
        __syncthreads();
        uint32_t* tmp = src; src = dst; dst = tmp;
    }
    uint32_t r = blockSums[blockIdx.x] + src[t] - s;  // grid-exclusive rank
#pragma unroll
    for (int e = 0; e < SCAN_ELEMS; ++e) {
        rank[base + e] = r;
        r += occ[e];
    }
}

// --- 4. per-voxel outputs: features_avg and first_pos ----------------------
__global__ __launch_bounds__(TPB) void k_finalize(const uint32_t* __restrict__ count,
                                                  const uint32_t* __restrict__ firstIdx,
                                                  const uint32_t* __restrict__ rank,
                                                  const float* __restrict__ sums,
                                                  const float* __restrict__ pos,
                                                  float* __restrict__ out, int V) {
    uint32_t k = blockIdx.x * TPB + threadIdx.x;
    if (k >= NKEYS) return;
    uint32_t c = count[k];
    if (c == 0u) return;
    uint32_t v = rank[k];
    float inv = 1.0f / (float)c;
    out[3u * v + 0u] = sums[3u * k + 0u] * inv;
    out[3u * v + 1u] = sums[3u * k + 1u] * inv;
    out[3u * v + 2u] = sums[3u * k + 2u] * inv;
    size_t fb = (size_t)firstIdx[k] * 3u;
    float* fp = out + (size_t)3u * (size_t)V;
    fp[3u * v + 0u] = pos[fb + 0];
    fp[3u * v + 1u] = pos[fb + 1];
    fp[3u * v + 2u] = pos[fb + 2];
}

// --- 5a. voxel id per point via cached keys (32 MB read) -------------------
__global__ __launch_bounds__(TPB) void k_vox_keys(const uint32_t* __restrict__ keys,
                                                  const uint32_t* __restrict__ rank,
                                                  float* __restrict__ outv, int n) {
    int i = blockIdx.x * TPB + threadIdx.x;
    if (i >= n) return;
    __builtin_prefetch(keys + i + 16384, 0, 3);
    outv[i] = (float)rank[keys[i]];    // ids < 2^21 are exact in f32
}

// --- 5b. fallback: recompute key from positions (96 MB read) ---------------
__global__ __launch_bounds__(TPB) void k_vox_pos(const float* __restrict__ pos,
                                                 const uint32_t* __restrict__ rank,
                                                 float* __restrict__ outv, int n) {
    int i = blockIdx.x * TPB + threadIdx.x;
    if (i >= n) return;
    size_t b = (size_t)i * 3u;
    __builtin_prefetch(pos + b + 3u * 8192u, 0, 3);
    uint32_t key = voxel_key(pos[b + 0], pos[b + 1], pos[b + 2]);
    outv[i] = (float)rank[key];
}

// ---------------------------------------------------------------------------
extern "C" void kernel_launch(void* const* d_in, const int* in_sizes, int n_in,
                              void* d_out, int out_size, void* d_ws, size_t ws_size,
                              hipStream_t stream) {
    const float* feat = (const float*)d_in[0];   // [N,3]
    const float* pos  = (const float*)d_in[1];   // [N,3]
    int n = in_sizes[0] / 3;
    int V = (out_size - n) / 6;                  // occupied voxel count from harness

    // workspace layout (~50.3 MB core, +4*N optional key cache)
    uint32_t* count    = (uint32_t*)d_ws;                       // NKEYS
    uint32_t* firstIdx = count + NKEYS;                         // NKEYS
    uint32_t* rank     = firstIdx + NKEYS;                      // NKEYS
    float*    sums     = (float*)(rank + NKEYS);                // 3*NKEYS
    uint32_t* blockSums= (uint32_t*)(sums + (size_t)3 * NKEYS); // SCAN_NBLK
    uint32_t* keyCache = blockSums + SCAN_NBLK;                 // N (optional)
    size_t core_bytes  = (size_t)(6 * NKEYS + SCAN_NBLK) * 4u;
    bool use_keys      = ws_size >= core_bytes + (size_t)n * 4u;  // fixed per harness
    float* out         = (float*)d_out;

    int nblk   = (n + TPB - 1) / TPB;
    int ntiles = (n + ACC_TPB - 1) / ACC_TPB;
    int accgrid = ntiles < ACC_GRID ? ntiles : ACC_GRID;

    k_init    <<<NKEYS / TPB, TPB, 0, stream>>>(count, firstIdx, sums);
    k_accum   <<<accgrid, ACC_TPB, 0, stream>>>(pos, feat, n, ntiles, count, firstIdx,
                                                sums, use_keys ? keyCache : nullptr);
    k_scan1   <<<SCAN_NBLK, TPB, 0, stream>>>(count, blockSums);
    k_scan2   <<<1, SCAN_NBLK, 0, stream>>>(blockSums);
    k_scan3   <<<SCAN_NBLK, TPB, 0, stream>>>(count, blockSums, rank);
    k_finalize<<<NKEYS / TPB, TPB, 0, stream>>>(count, firstIdx, rank, sums, pos, out, V);
    if (use_keys)
        k_vox_keys<<<nblk, TPB, 0, stream>>>(keyCache, rank, out + (size_t)6 * (size_t)V, n);
    else
        k_vox_pos <<<nblk, TPB, 0, stream>>>(pos, rank, out + (size_t)6 * (size_t)V, n);
}